// AE_spikes_43928925504088
// MI455X (gfx1250) — compile-verified
//
#include <hip/hip_runtime.h>
#include <hip/hip_bf16.h>

// SNN autoencoder forward, 16 steps, batched across 256 images.
// Design: 16 workgroups x 256 threads (8 wave32). Each WG owns 16 images
// (the WMMA N dimension). All recurrent "totals" state lives in registers as
// WMMA C/D tiles; per-step binary spike matrices live in LDS. All GEMMs use
// exact fp32 WMMA (V_WMMA_F32_16X16X4_F32) since spikes are exactly {0,1}
// and weights are fp32 -> bit-accurate accumulation on the matrix pipe.

typedef __attribute__((ext_vector_type(2))) float v2f;
typedef __attribute__((ext_vector_type(8))) float v8f;

#define IN_N  784
#define HID   128
#define NSTEP 16
#define COLS  16   // images per workgroup (WMMA N)

__device__ __forceinline__ v8f wmma_k4(v2f a, v2f b, v8f c) {
  // 8 args: (neg_a, A, neg_b, B, c_mod, C, reuse_a, reuse_b)
  return __builtin_amdgcn_wmma_f32_16x16x4_f32(
      false, a, false, b, (short)0, c, false, false);
}

// acc(16x16, rows row0..row0+15, cols = 16 images) += W[row0:,:Kdim] * S^T
// W is row-major [M][Kdim] (K fast axis -> contiguous float2 loads).
// S is LDS, image-major [16][Kdim]   (K fast axis -> contiguous ds_load_b64).
// A frag: M = lane&15, K = v + 2*(lane>>4) + 4k
// B frag: N = lane&15, K = v + 2*(lane>>4) + 4k
__device__ __forceinline__ v8f gemm_tile(const float* __restrict__ W,
                                         const float* __restrict__ S,
                                         int row0, int Kdim, int lane, v8f acc) {
  const int hl = lane >> 4;
  const int ln = lane & 15;
  const float* wrow = W + (size_t)(row0 + ln) * Kdim + 2 * hl;
  const float* srow = S + ln * Kdim + 2 * hl;
#pragma unroll 4
  for (int k = 0; k < Kdim; k += 4) {
    v2f a = *(const v2f*)(wrow + k);
    v2f b = *(const v2f*)(srow + k);
    acc = wmma_k4(a, b, acc);
  }
  return acc;
}

__global__ __launch_bounds__(256) void snn_ae_kernel(
    const float* __restrict__ feat,
    const float* __restrict__ W0, const float* __restrict__ b0,
    const float* __restrict__ W1, const float* __restrict__ b1,
    const float* __restrict__ W2, const float* __restrict__ b2,
    const float* __restrict__ W3, const float* __restrict__ b3,
    float* __restrict__ out) {
  extern __shared__ float smem[];
  float* S_in = smem;                  // [COLS][IN_N] input-layer spikes
  float* Tin  = S_in + COLS * IN_N;    // [COLS][IN_N] input-layer totals
  float* S0   = Tin  + COLS * IN_N;    // [COLS][HID]
  float* S1   = S0   + COLS * HID;     // [COLS][HID]
  float* S2   = S1   + COLS * HID;     // [COLS][HID]

  const int tid  = threadIdx.x;
  const int wave = tid >> 5;
  const int lane = tid & 31;
  const int hl   = lane >> 4;   // half-lane select (C/D rows +8, A/B K +2)
  const int ln   = lane & 15;
  const int wg   = blockIdx.x;

  // elementwise mapping for the 784x16 input layer: 49 elems / thread
  const int ec = tid >> 4;   // image column 0..15
  const int en = tid & 15;   // neuron index base (conflict-free LDS stride)

  // input activation = floor(f*16)/16 (digitize against linspace(0,1,17))
  float act[49];
  {
    const float* f = feat + (size_t)(wg * COLS + ec) * IN_N;
#pragma unroll
    for (int j = 0; j < 49; ++j) {
      const int n = en + 16 * j;
      act[j] = floorf(f[n] * 16.0f) * 0.0625f;
      Tin[ec * IN_N + n] = 0.0f;
    }
  }

  // hidden-layer totals, one 16x16 C/D tile per wave, initialized to bias
  // (bin scale s = 1, so raw weights/biases). C/D: N = ln, M = v + 8*hl.
  v8f t0, t1, t2;
#pragma unroll
  for (int v = 0; v < 8; ++v) {
    const int m = wave * 16 + v + 8 * hl;
    t0[v] = b0[m];
    t1[v] = b1[m];
    t2[v] = b2[m];
  }

  // output-layer totals + spike accumulators: 49 M-tiles over 8 waves
  const int nt3 = (wave == 0) ? 7 : 6;   // 49 = 7 + 6*7 (wave-uniform)
  v8f t3[7], oac[7];
#pragma unroll
  for (int q = 0; q < 7; ++q)
#pragma unroll
    for (int v = 0; v < 8; ++v) { t3[q][v] = 0.0f; oac[q][v] = 0.0f; }
  for (int q = 0; q < nt3; ++q) {
    const int row0 = (wave + 8 * q) * 16;
#pragma unroll
    for (int v = 0; v < 8; ++v) t3[q][v] = b3[row0 + v + 8 * hl];
  }

  __syncthreads();

  for (int step = 0; step < NSTEP; ++step) {
    // ---- input layer fire (elementwise) ----
#pragma unroll 7
    for (int j = 0; j < 49; ++j) {
      const int idx = ec * IN_N + en + 16 * j;
      const float t = Tin[idx] + act[j];
      const float s = (t > 1.0f) ? 1.0f : 0.0f;
      Tin[idx]  = t - s;
      S_in[idx] = s;
    }
    __syncthreads();

    // ---- layer 0: 128x784 @ 784x16, one tile per wave ----
    {
      v8f acc = gemm_tile(W0, S_in, wave * 16, IN_N, lane, t0);
#pragma unroll
      for (int v = 0; v < 8; ++v) {
        const float s = (acc[v] > 1.0f) ? 1.0f : 0.0f;
        t0[v] = acc[v] - s;
        S0[ln * HID + wave * 16 + v + 8 * hl] = s;
      }
    }
    __syncthreads();

    // ---- layer 1: 128x128 ----
    {
      v8f acc = gemm_tile(W1, S0, wave * 16, HID, lane, t1);
#pragma unroll
      for (int v = 0; v < 8; ++v) {
        const float s = (acc[v] > 1.0f) ? 1.0f : 0.0f;
        t1[v] = acc[v] - s;
        S1[ln * HID + wave * 16 + v + 8 * hl] = s;
      }
    }
    __syncthreads();

    // ---- layer 2: 128x128 ----
    {
      v8f acc = gemm_tile(W2, S1, wave * 16, HID, lane, t2);
#pragma unroll
      for (int v = 0; v < 8; ++v) {
        const float s = (acc[v] > 1.0f) ? 1.0f : 0.0f;
        t2[v] = acc[v] - s;
        S2[ln * HID + wave * 16 + v + 8 * hl] = s;
      }
    }
    __syncthreads();

    // ---- layer 3: 784x128, 6-7 tiles per wave; accumulate output spikes ----
    for (int q = 0; q < nt3; ++q) {
      const int row0 = (wave + 8 * q) * 16;
      v8f acc = gemm_tile(W3, S2, row0, HID, lane, t3[q]);
#pragma unroll
      for (int v = 0; v < 8; ++v) {
        const float s = (acc[v] > 1.0f) ? 1.0f : 0.0f;
        t3[q][v] = acc[v] - s;
        oac[q][v] += s;
      }
    }
    __syncthreads();
  }

  // ---- epilogue: out = spike_count * (1/16) ----
  for (int q = 0; q < nt3; ++q) {
    const int row0 = (wave + 8 * q) * 16;
#pragma unroll
    for (int v = 0; v < 8; ++v) {
      const int n = row0 + v + 8 * hl;
      out[(size_t)(wg * COLS + ln) * IN_N + n] = oac[q][v] * 0.0625f;
    }
  }
}

extern "C" void kernel_launch(void* const* d_in, const int* in_sizes, int n_in,
                              void* d_out, int out_size, void* d_ws, size_t ws_size,
                              hipStream_t stream) {
  const float* feat = (const float*)d_in[0];
  const float* W0   = (const float*)d_in[1];
  const float* b0   = (const float*)d_in[2];
  const float* W1   = (const float*)d_in[3];
  const float* b1   = (const float*)d_in[4];
  const float* W2   = (const float*)d_in[5];
  const float* b2   = (const float*)d_in[6];
  const float* W3   = (const float*)d_in[7];
  const float* b3   = (const float*)d_in[8];
  float* out = (float*)d_out;

  const size_t smemBytes = (size_t)(2 * COLS * IN_N + 3 * COLS * HID) * sizeof(float);
  snn_ae_kernel<<<16, 256, smemBytes, stream>>>(feat, W0, b0, W1, b1, W2, b2,
                                                W3, b3, out);
}